// DiscriminativeLoss_23587960389730
// MI455X (gfx1250) — compile-verified
//
#include <hip/hip_runtime.h>
#include <cmath>

#define NPIX   (512 * 512)
#define NBATCH 4
#define NCH    16
#define KCL    8

typedef __attribute__((ext_vector_type(2))) float v2f;
typedef __attribute__((ext_vector_type(8))) float v8f;

// ---------------------------------------------------------------------------
// Kernel 0: zero the accumulator workspace and the output scalar.
// ---------------------------------------------------------------------------
__global__ void dl_zero_kernel(float* __restrict__ ws_sums, float* __restrict__ out) {
    int t = threadIdx.x;
    if (t < NBATCH * KCL * NCH) ws_sums[t] = 0.0f;
    if (t == 0) out[0] = 0.0f;
}

// ---------------------------------------------------------------------------
// Kernel 1: per-cluster channel sums via V_WMMA_F32_16X16X4_F32.
//   D(16x16) += A(16x4 one-hot of labels) * B(4 pixels x 16 channels)
// A lanes 0-15: (M=lane, K=0..1); lanes 16-31: (M=lane-16, K=2..3)
// B VGPR v, lanes 0-15: (K=v, N=lane); lanes 16-31: (K=2+v, N=lane-16)
// Two independent accumulator chains per wave so consecutive WMMAs are not
// serially dependent and 4 loads stay in flight per iteration.
// ---------------------------------------------------------------------------
__global__ __launch_bounds__(256) void dl_sums_kernel(
        const float* __restrict__ data,     // (B, 16, NPIX)
        const int*   __restrict__ labels,   // (B, NPIX)
        float*       __restrict__ ws_sums)  // (B, 8, 16)
{
    const int b     = blockIdx.y;
    const int lane  = threadIdx.x & 31;
    const int wave  = threadIdx.x >> 5;            // 0..7
    const int gwave = blockIdx.x * 8 + wave;       // global wave id in batch
    const int nwaves = gridDim.x * 8;
    const int half  = lane >> 4;                   // 0 -> K 0,1 ; 1 -> K 2,3
    const int mn    = lane & 15;                   // A row (cluster) / B col (channel)

    const int nchunks = NPIX / 4;                  // 4 pixels per WMMA
    const int per   = (nchunks + nwaves - 1) / nwaves;
    const int start = gwave * per;
    const int end   = (start + per < nchunks) ? (start + per) : nchunks;
    const int cnt   = end - start;

    // Per-lane streaming pointers (advance 4 pixels per chunk).
    const float* dp = data + ((size_t)b * NCH + mn) * NPIX + (size_t)start * 4 + half * 2;
    const int*   lp = labels + (size_t)b * NPIX + (size_t)start * 4 + half * 2;

    v8f acc0 = {};
    v8f acc1 = {};
    int it = 0;
    for (; it + 1 < cnt; it += 2) {
        // chunk it -> acc0
        v2f b0 = *(const v2f*)dp;
        int l0 = lp[0];
        int l1 = lp[1];
        // chunk it+1 -> acc1
        v2f b1 = *(const v2f*)(dp + 4);
        int l2 = lp[4];
        int l3 = lp[5];

        v2f a0, a1;
        a0.x = (l0 == mn) ? 1.0f : 0.0f;
        a0.y = (l1 == mn) ? 1.0f : 0.0f;
        a1.x = (l2 == mn) ? 1.0f : 0.0f;
        a1.y = (l3 == mn) ? 1.0f : 0.0f;

        acc0 = __builtin_amdgcn_wmma_f32_16x16x4_f32(
                   false, a0, false, b0, (short)0, acc0, false, false);
        acc1 = __builtin_amdgcn_wmma_f32_16x16x4_f32(
                   false, a1, false, b1, (short)0, acc1, false, false);

        dp += 8;
        lp += 8;
    }
    if (it < cnt) {                                // remainder (not hit when
        v2f b0 = *(const v2f*)dp;                  //  nchunks % nwaves == 0)
        int l0 = lp[0];
        int l1 = lp[1];
        v2f a0;
        a0.x = (l0 == mn) ? 1.0f : 0.0f;
        a0.y = (l1 == mn) ? 1.0f : 0.0f;
        acc0 = __builtin_amdgcn_wmma_f32_16x16x4_f32(
                   false, a0, false, b0, (short)0, acc0, false, false);
    }

    v8f acc = acc0 + acc1;

    // lanes 0..15 hold clusters 0..7 (VGPR v = cluster v), channel = lane.
    if (half == 0) {
        float* s = ws_sums + (size_t)b * KCL * NCH + mn;
#pragma unroll
        for (int v = 0; v < KCL; ++v) {
            atomicAdd(s + v * NCH, acc[v]);
        }
    }
}

// ---------------------------------------------------------------------------
// Kernel 2: mu = sums/n ; pairwise center hinge term (includes self term, as
// the reference does).  One block of 512 threads.
// ---------------------------------------------------------------------------
__global__ __launch_bounds__(512) void dl_mu_dist_kernel(
        const float* __restrict__ ws_sums,
        float*       __restrict__ ws_mu,
        float*       __restrict__ out)
{
    __shared__ float smu[NBATCH * KCL * NCH];      // 512 floats
    const int t = threadIdx.x;

    float m = ws_sums[t] * (1.0f / (float)NPIX);
    smu[t]   = m;
    ws_mu[t] = m;
    __syncthreads();

    if (t < NBATCH * KCL * KCL) {                  // 256 (b, i, j) triples
        const int b = t >> 6;
        const int i = (t >> 3) & 7;
        const int j = t & 7;
        const float* mi = smu + (b * KCL + i) * NCH;
        const float* mj = smu + (b * KCL + j) * NCH;
        float d2 = 0.0f;
#pragma unroll
        for (int c = 0; c < NCH; ++c) {
            float df = mi[c] - mj[c];
            d2 += df * df;
        }
        float pd = sqrtf(d2);
        float h  = fmaxf(2.0f - pd, 0.0f);         // DELTA_DIST = 2
        atomicAdd(out, h * h * (1.0f / 14.0f));    // /(K-1)/2
    }
}

// ---------------------------------------------------------------------------
// Kernel 3: variance hinge pass.  Sum over clusters of vars_k collapses to a
// single global sum of per-pixel hinges / n.
// ---------------------------------------------------------------------------
__global__ __launch_bounds__(256) void dl_var_kernel(
        const float* __restrict__ data,
        const int*   __restrict__ labels,
        const float* __restrict__ ws_mu,
        float*       __restrict__ out)
{
    const int b = blockIdx.y;

    __shared__ float smu[KCL * NCH];               // 128 floats
    if (threadIdx.x < KCL * NCH)
        smu[threadIdx.x] = ws_mu[(size_t)b * KCL * NCH + threadIdx.x];
    __syncthreads();

    const float* dbase = data + (size_t)b * NCH * NPIX;
    const int*   lab   = labels + (size_t)b * NPIX;

    const int tid    = blockIdx.x * blockDim.x + threadIdx.x;
    const int stride = gridDim.x * blockDim.x;

    float local = 0.0f;
    for (int p = tid; p < NPIX; p += stride) {
        const int l = lab[p];
        const float* mrow = smu + l * NCH;
        float d2 = 0.0f;
#pragma unroll
        for (int c = 0; c < NCH; ++c) {
            float x  = dbase[(size_t)c * NPIX + p];   // coalesced per plane
            float df = x - mrow[c];
            d2 += df * df;
        }
        float nr = sqrtf(d2);
        float h  = fmaxf(nr - 1.0f, 0.0f);         // DELTA_VAR = 1
        local += h * h;
    }

    __shared__ float red[256];
    red[threadIdx.x] = local;
    __syncthreads();
#pragma unroll
    for (int s = 128; s > 0; s >>= 1) {
        if (threadIdx.x < s) red[threadIdx.x] += red[threadIdx.x + s];
        __syncthreads();
    }
    if (threadIdx.x == 0)
        atomicAdd(out, red[0] * (1.0f / (float)NPIX));
}

// ---------------------------------------------------------------------------
extern "C" void kernel_launch(void* const* d_in, const int* in_sizes, int n_in,
                              void* d_out, int out_size, void* d_ws, size_t ws_size,
                              hipStream_t stream) {
    const float* data   = (const float*)d_in[0];   // (4,16,512,512) f32
    const int*   labels = (const int*)d_in[1];     // (4,512,512) i32
    float* out = (float*)d_out;                    // scalar f32 loss

    float* ws_sums = (float*)d_ws;                 // 512 floats
    float* ws_mu   = ws_sums + NBATCH * KCL * NCH; // 512 floats

    dl_zero_kernel<<<1, 512, 0, stream>>>(ws_sums, out);

    dim3 g1(256, NBATCH);                          // 2048 waves/batch -> 32 chunks each
    dl_sums_kernel<<<g1, 256, 0, stream>>>(data, labels, ws_sums);

    dl_mu_dist_kernel<<<1, 512, 0, stream>>>(ws_sums, ws_mu, out);

    dim3 g2(256, NBATCH);
    dl_var_kernel<<<g2, 256, 0, stream>>>(data, labels, ws_mu, out);
}